// SubpixelCrossEntropy_3719441678343
// MI455X (gfx1250) — compile-verified
//
#include <hip/hip_runtime.h>

// SubpixelCrossEntropy for MI455X (gfx1250).
// Memory-bound: 512MB similarities streamed once -> ~22us floor at 23.3 TB/s.
// CDNA5 path: async global->LDS copies (ASYNCcnt), triple-buffered, with
// non-temporal hints so the single-use stream doesn't thrash the 192MB L2.

constexpr int Dn = 64;
constexpr int Hn = 256;
constexpr int Wn = 512;
constexpr int HW = Hn * Wn;          // 131072 (2^17)
constexpr int TPB = 256;             // 8 wave32 per block
constexpr int CHUNK = 4;             // d-planes per async chunk = 4KB per block
constexpr int NCHUNK = Dn / CHUNK;   // 16
constexpr int NBUF = 3;              // triple buffer -> 2 chunks in flight

__device__ __forceinline__ void async_chunk_to_lds(const float* gbase,
                                                   unsigned voff,
                                                   unsigned ldsaddr) {
  // GVS mode: mem = SADDR(64b, uniform) + VADDR(i32, per-lane); each lane 16B.
  // NT: single-use stream, keep it out of cache residency.
  asm volatile("global_load_async_to_lds_b128 %0, %1, %2 th:TH_LOAD_NT"
               :
               : "v"(ldsaddr), "v"(voff), "s"(gbase)
               : "memory");
}

__device__ __forceinline__ void wait_async_le(int n) {
  // ASYNCcnt decrements in issue order -> wait<=1 with 2 outstanding means
  // the oldest transfer has fully landed in LDS.
  if (n == 0)
    asm volatile("s_wait_asynccnt 0x0" ::: "memory");
  else
    asm volatile("s_wait_asynccnt 0x1" ::: "memory");
}

__global__ __launch_bounds__(TPB) void sce_main(const float* __restrict__ sim,
                                                const float* __restrict__ gt,
                                                float* __restrict__ partials,
                                                int nblocks) {
  __shared__ float buf[NBUF][CHUNK][TPB];       // 12KB triple buffer
  __shared__ float wsum[TPB / 32], wcnt[TPB / 32];

  const int t = threadIdx.x;
  const long long P0 = (long long)blockIdx.x * TPB;   // first pixel of block
  const int bb = (int)(P0 >> 17);                     // / (H*W)
  const int hw = (int)(P0 & (HW - 1));
  const float* gbase = sim + (size_t)bb * Dn * HW + hw;  // (b, d=0, h, w0)

  // Lane's slice of a chunk: plane q (0..3) within chunk, 16B unit j (0..63).
  const int q = t >> 6;
  const int j = t & 63;
  const unsigned voff0 = (unsigned)(q * HW * 4 + j * 16);
  const unsigned chunkStride = (unsigned)(CHUNK * HW * 4);   // 2MB per chunk step
  const unsigned bufStride = (unsigned)(CHUNK * TPB * 4);    // 4KB per buffer
  const unsigned ldsBase = (unsigned)(uintptr_t)(&buf[0][0][0]);
  const unsigned ldsLane = ldsBase + (unsigned)(q * (TPB * 4) + j * 16);

  const float g = __builtin_nontemporal_load(gt + P0 + t);
  const bool known = __builtin_isfinite(g) != 0;
  const float gs = known ? g : 0.0f;

  float m = -__builtin_huge_valf();   // online logsumexp state
  float se = 0.0f;
  float sumW = 0.0f;
  float sumWX = 0.0f;

  // Prologue: two chunks in flight.
  async_chunk_to_lds(gbase, voff0, ldsLane);                               // c=0
  async_chunk_to_lds(gbase, voff0 + chunkStride, ldsLane + bufStride);     // c=1

  for (int c = 0; c < NCHUNK; ++c) {
    wait_async_le((c + 1 < NCHUNK) ? 1 : 0);  // chunk c landed (this wave)
    __syncthreads();                          // ...for all waves; buf[(c-1)%3] free
    if (c + 2 < NCHUNK) {
      async_chunk_to_lds(gbase, voff0 + (unsigned)(c + 2) * chunkStride,
                         ldsLane + (unsigned)((c + 2) % NBUF) * bufStride);
    }
    const float* bp = &buf[c % NBUF][0][t];
#pragma unroll
    for (int k = 0; k < CHUNK; ++k) {
      const float x = bp[k * TPB];
      const float disp = (float)(2 * (c * CHUNK + k));   // DISPARITY_STEP=2
      const float w = __expf(-fabsf(gs - disp));         // DIVERSITY=1
      const float nm = fmaxf(m, x);
      se = se * __expf(m - nm) + __expf(x - nm);
      m = nm;
      sumW += w;
      sumWX = fmaf(w, x, sumWX);
    }
  }

  const float logZ = m + __logf(se);
  const float ent = logZ - sumWX / sumW;   // = -sum(Pt*logP)/sum(Pt)
  float contrib = known ? ent : 0.0f;
  float cnt = known ? 1.0f : 0.0f;

  // wave32 tree reduction
  for (int off = 16; off > 0; off >>= 1) {
    contrib += __shfl_down(contrib, off);
    cnt += __shfl_down(cnt, off);
  }
  const int wave = t >> 5;
  if ((t & 31) == 0) { wsum[wave] = contrib; wcnt[wave] = cnt; }
  __syncthreads();
  if (t == 0) {
    float S = 0.0f, C = 0.0f;
#pragma unroll
    for (int i = 0; i < TPB / 32; ++i) { S += wsum[i]; C += wcnt[i]; }
    partials[blockIdx.x] = S;                 // masked entropy sum
    partials[nblocks + blockIdx.x] = C;       // known count
  }
}

__global__ __launch_bounds__(TPB) void sce_finalize(const float* __restrict__ partials,
                                                    float* __restrict__ out,
                                                    int nblocks) {
  __shared__ float ss[TPB], cc[TPB];
  const int t = threadIdx.x;
  float S = 0.0f, C = 0.0f;
  for (int i = t; i < nblocks; i += TPB) {
    S += partials[i];
    C += partials[nblocks + i];
  }
  ss[t] = S; cc[t] = C;
  __syncthreads();
  for (int s2 = TPB / 2; s2 > 0; s2 >>= 1) {
    if (t < s2) { ss[t] += ss[t + s2]; cc[t] += cc[t + s2]; }
    __syncthreads();
  }
  if (t == 0) out[0] = ss[0] / cc[0];
}

extern "C" void kernel_launch(void* const* d_in, const int* in_sizes, int n_in,
                              void* d_out, int out_size, void* d_ws, size_t ws_size,
                              hipStream_t stream) {
  (void)n_in; (void)out_size; (void)ws_size;
  const float* sim = (const float*)d_in[0];   // [B, D, H, W] fp32
  const float* gt  = (const float*)d_in[1];   // [B, H, W] fp32 (inf = unknown)
  float* out = (float*)d_out;                 // scalar fp32
  float* partials = (float*)d_ws;             // nblocks sums + nblocks counts (32KB)

  const int npix = in_sizes[1];               // B*H*W = 1,048,576
  const int nblocks = npix / TPB;             // 4096

  sce_main<<<nblocks, TPB, 0, stream>>>(sim, gt, partials, nblocks);
  sce_finalize<<<1, TPB, 0, stream>>>(partials, out, nblocks);
}